// BiCG_Net_87883620811277
// MI455X (gfx1250) — compile-verified
//
#include <hip/hip_runtime.h>
#include <math.h>

// ---------------- geometry ----------------
#define N      512
#define NN     (N * N)
#define NCH    8
#define PBLK   256                 // partial blocks per channel (matches 256-way WMMA reduce)
#define OUTN   514
#define OUT2   (OUTN * OUTN)
#define THRESH (1e-9f * 262144.0f) // EPS * nx * ny
#define EPSF   1e-9f

// ---------------- per-channel scalar slots (each slot = NCH floats) ----------------
#define SC_MEAN  0
#define SC_RABS  1
#define SC_R0ABS 2
#define SC_RHO   3
#define SC_SIGMA 4
#define SC_ALPHA 5
#define SC_OMEGA 6
#define SC_BETA  7
#define SC_VABS  8
#define SC_FROBS 9
#define SC_K     10
#define SC_RES1  11
#define SC_NRES  12
#define SC_C3    13
#define SC_C4    14
#define SC_COUNT 16

typedef float v2f __attribute__((ext_vector_type(2)));
typedef float v8f __attribute__((ext_vector_type(8)));
typedef int vint4 __attribute__((vector_size(16)));   // matches builtin's expected param type

#define AS3 __attribute__((address_space(3)))

// ---------------- CDNA5 feature detection (compile-safe) ----------------
#if defined(__has_builtin)
#  if __has_builtin(__builtin_amdgcn_global_load_async_to_lds_b128) && \
      __has_builtin(__builtin_amdgcn_s_wait_asynccnt)
#    define USE_ASYNC_LDS 1
#  endif
#  if __has_builtin(__builtin_amdgcn_wmma_f32_16x16x4_f32)
#    define USE_WMMA 1
#  endif
#endif
#ifndef USE_ASYNC_LDS
#  define USE_ASYNC_LDS 0
#endif
#ifndef USE_WMMA
#  define USE_WMMA 0
#endif

// ---------------- small helpers ----------------
__device__ __forceinline__ float4 ld4(const float* p) { return *reinterpret_cast<const float4*>(p); }
__device__ __forceinline__ void   st4(float* p, float4 v) { *reinterpret_cast<float4*>(p) = v; }

// Sum 256 floats with one wave via V_WMMA_F32_16X16X4_F32 against all-ones B.
// D[i][j] = sum_k A[i][k]; row sums accumulated over 4 chunks; final cross-lane add.
// Returns the total in ALL lanes of the wave.
__device__ __forceinline__ float wave_sum256(const float* __restrict__ arr)
{
  const int lane = threadIdx.x & 31;
#if USE_WMMA
  const int base = ((lane & 15) << 2) + ((lane >> 4) << 1); // A 16x4 f32 layout
  v8f acc = {0.f, 0.f, 0.f, 0.f, 0.f, 0.f, 0.f, 0.f};
  v2f ones; ones[0] = 1.f; ones[1] = 1.f;
  #pragma unroll
  for (int c = 0; c < 4; ++c) {
    v2f a;
    a[0] = arr[c * 64 + base];
    a[1] = arr[c * 64 + base + 1];
    acc = __builtin_amdgcn_wmma_f32_16x16x4_f32(false, a, false, ones,
                                                (short)0, acc, false, false);
  }
  float s = acc[0] + acc[1] + acc[2] + acc[3] + acc[4] + acc[5] + acc[6] + acc[7];
  s += __shfl_xor(s, 16, 32);
  return s;
#else
  float s = 0.f;
  for (int i = lane; i < 256; i += 32) s += arr[i];
  #pragma unroll
  for (int o = 16; o >= 1; o >>= 1) s += __shfl_xor(s, o, 32);
  return s;
#endif
}

// block(256) -> two scalars into partial arrays at [slot]
__device__ __forceinline__ void block_store_partials(float a, float b,
    float* __restrict__ pA, float* __restrict__ pB, int slot)
{
  #pragma unroll
  for (int o = 16; o >= 1; o >>= 1) {
    a += __shfl_xor(a, o, 32);
    b += __shfl_xor(b, o, 32);
  }
  __shared__ float rA[8], rB[8];
  const int wid  = threadIdx.x >> 5;
  const int lane = threadIdx.x & 31;
  if (lane == 0) { rA[wid] = a; rB[wid] = b; }
  __syncthreads();
  if (threadIdx.x == 0) {
    float ta = 0.f, tb = 0.f;
    #pragma unroll
    for (int i = 0; i < 8; ++i) { ta += rA[i]; tb += rB[i]; }
    pA[slot] = ta;
    pB[slot] = tb;
  }
}

// ---------------- 5-point stencil on a 2-row tile with async-LDS halo ----------------
// Block handles rows [2*blk, 2*blk+1] of one channel; LDS holds 4 halo rows.
// Reflected (symmetric) boundaries: off-edge neighbor == edge value.
__device__ __forceinline__ void spmv_tile(
    const float* __restrict__ z,
    const float* __restrict__ boo, const float* __restrict__ bmo,
    const float* __restrict__ bom, const float* __restrict__ bop,
    const float* __restrict__ bpo,
    int blk, float out[4], float cen[4], int* gbase_out)
{
  __shared__ float lds[4 * N];
  const int R   = blk * 2;
  const int tid = threadIdx.x;

  #pragma unroll
  for (int q = 0; q < 2; ++q) {
    int v4i  = tid + q * 256;          // 0..511 float4 slots (4 rows * 128)
    int row  = v4i >> 7;               // halo row 0..3
    int col4 = v4i & 127;
    int srow = R - 1 + row;
    srow = srow < 0 ? 0 : (srow > N - 1 ? N - 1 : srow);
#if USE_ASYNC_LDS
    __builtin_amdgcn_global_load_async_to_lds_b128(
        (vint4*)(z + (size_t)srow * N + col4 * 4),
        (AS3 vint4*)(lds + row * N + col4 * 4), 0, 0);
#else
    st4(lds + row * N + col4 * 4, ld4(z + (size_t)srow * N + col4 * 4));
#endif
  }
#if USE_ASYNC_LDS
  __builtin_amdgcn_s_wait_asynccnt(0);
#endif
  __syncthreads();

  const int gbase = R * N + tid * 4;
  float4 f0 = ld4(boo + gbase), f1 = ld4(bmo + gbase), f2 = ld4(bom + gbase);
  float4 f3 = ld4(bop + gbase), f4 = ld4(bpo + gbase);
  float aboo[4] = {f0.x, f0.y, f0.z, f0.w};
  float abmo[4] = {f1.x, f1.y, f1.z, f1.w};
  float abom[4] = {f2.x, f2.y, f2.z, f2.w};
  float abop[4] = {f3.x, f3.y, f3.z, f3.w};
  float abpo[4] = {f4.x, f4.y, f4.z, f4.w};

  #pragma unroll
  for (int q = 0; q < 4; ++q) {
    int e   = tid * 4 + q;
    int lr  = 1 + (e >> 9);            // local LDS row 1 or 2
    int col = e & (N - 1);
    int cm  = col ? col - 1 : 0;
    int cp  = col < N - 1 ? col + 1 : N - 1;
    float c  = lds[lr * N + col];
    float up = lds[(lr - 1) * N + col];
    float dn = lds[(lr + 1) * N + col];
    float lf = lds[lr * N + cm];
    float rt = lds[lr * N + cp];
    // reference order: boo*c + bmo*up + bom*left + bop*right + bpo*down
    out[q] = aboo[q] * c + abmo[q] * up + abom[q] * lf + abop[q] * rt + abpo[q] * dn;
    cen[q] = c;
  }
  *gbase_out = gbase;
}

// =====================================================================
//                               KERNELS
// =====================================================================

// partial sums of V (for per-channel mean)
__global__ void __launch_bounds__(256) k_partial_sum_V(const float* __restrict__ V,
                                                       float* pA, float* pB)
{
  const int ch = blockIdx.y, blk = blockIdx.x;
  const size_t base = (size_t)ch * NN + blk * 1024 + threadIdx.x * 4;
  float4 v = ld4(V + base);
  block_store_partials(v.x + v.y + v.z + v.w, 0.f, pA, pB, ch * PBLK + blk);
}

__global__ void __launch_bounds__(256) k_finish_mean(float* S, const float* pA)
{
  const int w = threadIdx.x >> 5, lane = threadIdx.x & 31;
  const float s = wave_sum256(pA + w * PBLK);
  if (lane == 0) S[SC_MEAN * NCH + w] = s / 262144.0f;
}

// stencil coefficients (transposed, sym-padded V reads; one-time)
__global__ void __launch_bounds__(256) k_coeff(const float* __restrict__ V,
    const float* __restrict__ M1, const float* __restrict__ M2,
    float* boo, float* bmo, float* bom, float* bop, float* bpo)
{
  const int ch = blockIdx.y;
  const float* Vc  = V  + (size_t)ch * NN;
  const float* m1c = M1 + (size_t)ch * NN;
  const float* m2c = M2 + (size_t)ch * NN;
  const int e0 = blockIdx.x * 1024 + threadIdx.x * 4;
  #pragma unroll
  for (int q = 0; q < 4; ++q) {
    const int e = e0 + q;
    const int I = e >> 9, J = e & (N - 1);
    // sym(a): padded index a -> V index
    #define SYMI(a) ((a) - 1 < 0 ? 0 : ((a) - 1 > N - 1 ? N - 1 : (a) - 1))
    // Vt(i,j) = V[sym(j)][sym(i)]  (transposed)
    #define VT(i, j) Vc[(size_t)SYMI(j) * N + SYMI(i)]
    const float vA  = VT(I,     J + 1);
    const float vB  = VT(I + 1, J + 1);
    const float vC  = VT(I + 2, J + 1);
    const float vL  = VT(I + 1, J);
    const float vR  = VT(I + 1, J + 2);
    const float w1a = m1c[(size_t)SYMI(J + 1) * N + SYMI(I + 1)];
    const float w1b = m1c[(size_t)SYMI(J + 1) * N + SYMI(I)];
    const float w2a = m2c[(size_t)SYMI(J + 1) * N + SYMI(I + 1)];
    const float w2b = m2c[(size_t)SYMI(J)     * N + SYMI(I + 1)];
    const float da = w1a * (vC - vB) / (0.5f * (vC + vB)); // d1f[I+1][J+1]
    const float db = w1b * (vB - vA) / (0.5f * (vB + vA)); // d1f[I  ][J+1]
    const float dc = w2a * (vR - vB) / (0.5f * (vR + vB)); // d2f[I+1][J+1]
    const float dd = w2b * (vB - vL) / (0.5f * (vB + vL)); // d2f[I+1][J  ]
    const size_t o = (size_t)ch * NN + e;
    boo[o] = 401.0f + 50.0f * (da - db + dc - dd);
    bpo[o] = -100.0f + 50.0f * da;
    bmo[o] = -100.0f - 50.0f * db;
    bop[o] = -100.0f + 50.0f * dc;
    bom[o] = -100.0f - 50.0f * dd;
    #undef VT
    #undef SYMI
  }
}

// x = mean; rhs = mean - A(mean) = mean - mean*(sum coeffs); r=r0=p=rhs; v=s=t=0
__global__ void __launch_bounds__(256) k_init_fields(const float* S,
    const float* boo, const float* bmo, const float* bom, const float* bop, const float* bpo,
    float* x, float* r, float* r0, float* p, float* v, float* s, float* t,
    float* pA, float* pB)
{
  const int ch = blockIdx.y, blk = blockIdx.x;
  const float m = S[SC_MEAN * NCH + ch];
  const size_t base = (size_t)ch * NN + blk * 1024 + threadIdx.x * 4;
  float4 A = ld4(boo + base), B = ld4(bmo + base), C = ld4(bom + base);
  float4 D = ld4(bop + base), E = ld4(bpo + base);
  float4 rhs;
  rhs.x = m - (A.x * m + B.x * m + C.x * m + D.x * m + E.x * m);
  rhs.y = m - (A.y * m + B.y * m + C.y * m + D.y * m + E.y * m);
  rhs.z = m - (A.z * m + B.z * m + C.z * m + D.z * m + E.z * m);
  rhs.w = m - (A.w * m + B.w * m + C.w * m + D.w * m + E.w * m);
  float4 z4 = make_float4(0.f, 0.f, 0.f, 0.f);
  float4 m4 = make_float4(m, m, m, m);
  st4(x + base, m4);
  st4(r + base, rhs); st4(r0 + base, rhs); st4(p + base, rhs);
  st4(v + base, z4);  st4(s + base, z4);   st4(t + base, z4);
  const float a = rhs.x * rhs.x + rhs.y * rhs.y + rhs.z * rhs.z + rhs.w * rhs.w;
  block_store_partials(a, 0.f, pA, pB, ch * PBLK + blk);
}

__global__ void __launch_bounds__(256) k_init_reduce(float* S, const float* pA)
{
  const int w = threadIdx.x >> 5, lane = threadIdx.x & 31;
  const float ss = wave_sum256(pA + w * PBLK);
  if (lane == 0) {
    const float ra = sqrtf(ss);
    S[SC_RABS  * NCH + w] = ra;
    S[SC_R0ABS * NCH + w] = ra;
    S[SC_RHO   * NCH + w] = ss;   // dot(r, r0hat)
    S[SC_SIGMA * NCH + w] = 0.f;
    S[SC_ALPHA * NCH + w] = 0.f;
    S[SC_OMEGA * NCH + w] = 0.f;
    S[SC_BETA  * NCH + w] = 0.f;
    S[SC_VABS  * NCH + w] = 0.f;
    S[SC_FROBS * NCH + w] = 0.f;
    S[SC_K     * NCH + w] = 0.f;
    S[SC_RES1  * NCH + w] = 0.f;
    S[SC_NRES  * NCH + w] = 0.f;
    S[SC_C3    * NCH + w] = 0.f;
    S[SC_C4    * NCH + w] = 0.f;
  }
}

// v = A(p); partials: v·r0, v·v   (gated by conv = k<kmax && r_abs>thresh)
__global__ void __launch_bounds__(256) k_spmv_v(const float* S, const int* kmaxp,
    const float* boo, const float* bmo, const float* bom, const float* bop, const float* bpo,
    const float* p, const float* r0, float* v, float* pA, float* pB)
{
  const int ch = blockIdx.y, blk = blockIdx.x;
  if (!((S[SC_K * NCH + ch] < (float)(*kmaxp)) && (S[SC_RABS * NCH + ch] > THRESH))) return;
  const size_t cb = (size_t)ch * NN;
  float out[4], cen[4]; int gb;
  spmv_tile(p + cb, boo + cb, bmo + cb, bom + cb, bop + cb, bpo + cb, blk, out, cen, &gb);
  float4 w4 = ld4(r0 + cb + gb);
  float wa[4] = {w4.x, w4.y, w4.z, w4.w};
  float a = 0.f, b = 0.f;
  #pragma unroll
  for (int q = 0; q < 4; ++q) { a += out[q] * wa[q]; b += out[q] * out[q]; }
  st4(v + cb + gb, make_float4(out[0], out[1], out[2], out[3]));
  block_store_partials(a, b, pA, pB, ch * PBLK + blk);
}

__global__ void __launch_bounds__(256) k_reduce_sigma(float* S, const int* kmaxp,
                                                      const float* pA, const float* pB)
{
  const int w = threadIdx.x >> 5, lane = threadIdx.x & 31;
  const float sA = wave_sum256(pA + w * PBLK);  // dot(v, r0)
  const float sB = wave_sum256(pB + w * PBLK);  // dot(v, v)
  const bool conv = (S[SC_K * NCH + w] < (float)(*kmaxp)) && (S[SC_RABS * NCH + w] > THRESH);
  if (lane == 0) {
    float sigma = S[SC_SIGMA * NCH + w];
    float vabs  = S[SC_VABS  * NCH + w];
    if (conv) {
      sigma = sA;
      vabs  = sqrtf(sB);
      S[SC_SIGMA * NCH + w] = sigma;
      S[SC_VABS  * NCH + w] = vabs;
    }
    const bool res  = sigma <= EPSF * vabs * S[SC_R0ABS * NCH + w];
    const bool res1 = conv && res;
    const bool nres = conv && !res;
    S[SC_RES1 * NCH + w] = res1 ? 1.f : 0.f;
    S[SC_NRES * NCH + w] = nres ? 1.f : 0.f;
    if (nres) S[SC_ALPHA * NCH + w] = S[SC_RHO * NCH + w] / sigma;
  }
}

// restart: rhs = mean - A(x); p=r=r0=rhs; partial rhs^2  (gated res1)
__global__ void __launch_bounds__(256) k_spmv_restart(const float* S,
    const float* boo, const float* bmo, const float* bom, const float* bop, const float* bpo,
    const float* x, float* p, float* r, float* r0, float* pA, float* pB)
{
  const int ch = blockIdx.y, blk = blockIdx.x;
  if (S[SC_RES1 * NCH + ch] == 0.f) return;
  const float m = S[SC_MEAN * NCH + ch];
  const size_t cb = (size_t)ch * NN;
  float out[4], cen[4]; int gb;
  spmv_tile(x + cb, boo + cb, bmo + cb, bom + cb, bop + cb, bpo + cb, blk, out, cen, &gb);
  float rhs[4]; float a = 0.f;
  #pragma unroll
  for (int q = 0; q < 4; ++q) { rhs[q] = m - out[q]; a += rhs[q] * rhs[q]; }
  float4 o = make_float4(rhs[0], rhs[1], rhs[2], rhs[3]);
  st4(p + cb + gb, o); st4(r + cb + gb, o); st4(r0 + cb + gb, o);
  block_store_partials(a, 0.f, pA, pB, ch * PBLK + blk);
}

__global__ void __launch_bounds__(256) k_reduce_restart(float* S, const float* pA)
{
  const int w = threadIdx.x >> 5, lane = threadIdx.x & 31;
  const float ss = wave_sum256(pA + w * PBLK);
  if (lane == 0 && S[SC_RES1 * NCH + w] != 0.f) {
    const float ra = sqrtf(ss);
    S[SC_R0ABS * NCH + w] = ra;
    S[SC_RABS  * NCH + w] = ra;
    S[SC_RHO   * NCH + w] = ss;
    S[SC_K     * NCH + w] += 1.f;
  }
}

// s = r - alpha*v; partials: s·s, s·r0   (gated nres)
__global__ void __launch_bounds__(256) k_axpy_s(const float* S,
    const float* r, const float* v, const float* r0, float* s, float* pA, float* pB)
{
  const int ch = blockIdx.y, blk = blockIdx.x;
  if (S[SC_NRES * NCH + ch] == 0.f) return;
  const float alpha = S[SC_ALPHA * NCH + ch];
  const size_t base = (size_t)ch * NN + blk * 1024 + threadIdx.x * 4;
  float4 rr = ld4(r + base), vv = ld4(v + base), zz = ld4(r0 + base);
  float4 ss;
  ss.x = rr.x - alpha * vv.x; ss.y = rr.y - alpha * vv.y;
  ss.z = rr.z - alpha * vv.z; ss.w = rr.w - alpha * vv.w;
  st4(s + base, ss);
  const float a = ss.x * ss.x + ss.y * ss.y + ss.z * ss.z + ss.w * ss.w;
  const float b = ss.x * zz.x + ss.y * zz.y + ss.z * zz.z + ss.w * zz.w;
  block_store_partials(a, b, pA, pB, ch * PBLK + blk);
}

__global__ void __launch_bounds__(256) k_reduce_s(float* S, const float* pA, const float* pB)
{
  const int w = threadIdx.x >> 5, lane = threadIdx.x & 31;
  const float sA = wave_sum256(pA + w * PBLK);  // s·s
  const float sB = wave_sum256(pB + w * PBLK);  // s·r0
  if (lane == 0) {
    if (S[SC_NRES * NCH + w] != 0.f) {
      const float fs = sqrtf(sA);
      S[SC_FROBS * NCH + w] = fs;
      const bool conv2 = fs <= THRESH;
      S[SC_C3 * NCH + w] = conv2 ? 1.f : 0.f;
      S[SC_C4 * NCH + w] = conv2 ? 0.f : 1.f;
      if (conv2) {               // r will become s
        S[SC_RHO  * NCH + w] = sB;
        S[SC_RABS * NCH + w] = fs;
      }
      S[SC_K * NCH + w] += 1.f;
    } else {
      S[SC_C3 * NCH + w] = 0.f;
      S[SC_C4 * NCH + w] = 0.f;
    }
  }
}

// c3: x += alpha*p; r = s
__global__ void __launch_bounds__(256) k_update_c3(const float* S,
    const float* p, const float* s, float* x, float* r)
{
  const int ch = blockIdx.y, blk = blockIdx.x;
  if (S[SC_C3 * NCH + ch] == 0.f) return;
  const float alpha = S[SC_ALPHA * NCH + ch];
  const size_t base = (size_t)ch * NN + blk * 1024 + threadIdx.x * 4;
  float4 xx = ld4(x + base), pp = ld4(p + base), ss = ld4(s + base);
  xx.x += alpha * pp.x; xx.y += alpha * pp.y; xx.z += alpha * pp.z; xx.w += alpha * pp.w;
  st4(x + base, xx);
  st4(r + base, ss);
}

// t = A(s); partials: t·s, t·t   (gated c4)
__global__ void __launch_bounds__(256) k_spmv_t(const float* S,
    const float* boo, const float* bmo, const float* bom, const float* bop, const float* bpo,
    const float* s, float* t, float* pA, float* pB)
{
  const int ch = blockIdx.y, blk = blockIdx.x;
  if (S[SC_C4 * NCH + ch] == 0.f) return;
  const size_t cb = (size_t)ch * NN;
  float out[4], cen[4]; int gb;
  spmv_tile(s + cb, boo + cb, bmo + cb, bom + cb, bop + cb, bpo + cb, blk, out, cen, &gb);
  float a = 0.f, b = 0.f;
  #pragma unroll
  for (int q = 0; q < 4; ++q) { a += out[q] * cen[q]; b += out[q] * out[q]; }
  st4(t + cb + gb, make_float4(out[0], out[1], out[2], out[3]));
  block_store_partials(a, b, pA, pB, ch * PBLK + blk);
}

__global__ void __launch_bounds__(256) k_reduce_omega(float* S, const float* pA, const float* pB)
{
  const int w = threadIdx.x >> 5, lane = threadIdx.x & 31;
  const float sA = wave_sum256(pA + w * PBLK);  // t·s
  const float sB = wave_sum256(pB + w * PBLK);  // t·t
  if (lane == 0 && S[SC_C4 * NCH + w] != 0.f)
    S[SC_OMEGA * NCH + w] = sA / sB;
}

// c4: x += alpha*p + omega*s; r = s - omega*t; partials: r·r0, r·r
__global__ void __launch_bounds__(256) k_update_c4(const float* S,
    const float* p, const float* s, const float* t, const float* r0,
    float* x, float* r, float* pA, float* pB)
{
  const int ch = blockIdx.y, blk = blockIdx.x;
  if (S[SC_C4 * NCH + ch] == 0.f) return;
  const float alpha = S[SC_ALPHA * NCH + ch];
  const float omega = S[SC_OMEGA * NCH + ch];
  const size_t base = (size_t)ch * NN + blk * 1024 + threadIdx.x * 4;
  float4 xx = ld4(x + base), pp = ld4(p + base), ss = ld4(s + base);
  float4 tt = ld4(t + base), zz = ld4(r0 + base);
  xx.x = xx.x + alpha * pp.x + omega * ss.x;
  xx.y = xx.y + alpha * pp.y + omega * ss.y;
  xx.z = xx.z + alpha * pp.z + omega * ss.z;
  xx.w = xx.w + alpha * pp.w + omega * ss.w;
  float4 rn;
  rn.x = ss.x - omega * tt.x; rn.y = ss.y - omega * tt.y;
  rn.z = ss.z - omega * tt.z; rn.w = ss.w - omega * tt.w;
  st4(x + base, xx);
  st4(r + base, rn);
  const float a = rn.x * zz.x + rn.y * zz.y + rn.z * zz.z + rn.w * zz.w;
  const float b = rn.x * rn.x + rn.y * rn.y + rn.z * rn.z + rn.w * rn.w;
  block_store_partials(a, b, pA, pB, ch * PBLK + blk);
}

__global__ void __launch_bounds__(256) k_reduce_beta(float* S, const float* pA, const float* pB)
{
  const int w = threadIdx.x >> 5, lane = threadIdx.x & 31;
  const float sA = wave_sum256(pA + w * PBLK);  // r_new · r0
  const float sB = wave_sum256(pB + w * PBLK);  // r_new · r_new
  if (lane == 0 && S[SC_C4 * NCH + w] != 0.f) {
    const float alpha   = S[SC_ALPHA * NCH + w];
    const float omega   = S[SC_OMEGA * NCH + w];
    const float rho_old = S[SC_RHO   * NCH + w]; // = dot(r_old, r0)
    S[SC_BETA * NCH + w] = alpha / omega * (sA / rho_old);
    S[SC_RHO  * NCH + w] = sA;
    S[SC_RABS * NCH + w] = sqrtf(sB);
  }
}

// c4: p = r + beta*(p - omega*v)
__global__ void __launch_bounds__(256) k_update_p(const float* S,
    const float* r, const float* v, float* p)
{
  const int ch = blockIdx.y, blk = blockIdx.x;
  if (S[SC_C4 * NCH + ch] == 0.f) return;
  const float beta  = S[SC_BETA  * NCH + ch];
  const float omega = S[SC_OMEGA * NCH + ch];
  const size_t base = (size_t)ch * NN + blk * 1024 + threadIdx.x * 4;
  float4 rr = ld4(r + base), vv = ld4(v + base), pp = ld4(p + base);
  pp.x = rr.x + beta * (pp.x - omega * vv.x);
  pp.y = rr.y + beta * (pp.y - omega * vv.y);
  pp.z = rr.z + beta * (pp.z - omega * vv.z);
  pp.w = rr.w + beta * (pp.w - omega * vv.w);
  st4(p + base, pp);
}

// emit padded 514x514 output: border = mean, interior = x
__global__ void __launch_bounds__(256) k_output(const float* S, const float* x, float* out)
{
  const long long TOT = (long long)NCH * OUT2;
  long long id0 = (long long)blockIdx.x * 1024 + threadIdx.x * 4;
  #pragma unroll
  for (int q = 0; q < 4; ++q) {
    long long id = id0 + q;
    if (id >= TOT) return;
    int b   = (int)(id / OUT2);
    int rem = (int)(id % OUT2);
    int i   = rem / OUTN;
    int j   = rem % OUTN;
    float val;
    if (i == 0 || i == OUTN - 1 || j == 0 || j == OUTN - 1)
      val = S[SC_MEAN * NCH + b];
    else
      val = x[(size_t)b * NN + (size_t)(i - 1) * N + (j - 1)];
    out[id] = val;
  }
}

// =====================================================================
//                               HOST
// =====================================================================
extern "C" void kernel_launch(void* const* d_in, const int* in_sizes, int n_in,
                              void* d_out, int out_size, void* d_ws, size_t ws_size,
                              hipStream_t stream) {
  (void)in_sizes; (void)n_in; (void)out_size; (void)ws_size;
  const float* V     = (const float*)d_in[0];
  const float* M1    = (const float*)d_in[1];
  const float* M2    = (const float*)d_in[2];
  const int*   kmaxp = (const int*)d_in[3];
  float* outp = (float*)d_out;

  float* ws = (float*)d_ws;
  const size_t F = (size_t)NCH * NN;      // 2,097,152 floats per field
  float* boo = ws + 0 * F;
  float* bmo = ws + 1 * F;
  float* bom = ws + 2 * F;
  float* bop = ws + 3 * F;
  float* bpo = ws + 4 * F;
  float* x   = ws + 5 * F;
  float* r   = ws + 6 * F;
  float* r0  = ws + 7 * F;
  float* p   = ws + 8 * F;
  float* v   = ws + 9 * F;
  float* s   = ws + 10 * F;
  float* t   = ws + 11 * F;
  float* pA  = ws + 12 * F;                       // NCH*PBLK
  float* pB  = pA + (size_t)NCH * PBLK;           // NCH*PBLK
  float* S   = pB + (size_t)NCH * PBLK;           // SC_COUNT*NCH

  const dim3 g(PBLK, NCH), blk(256);

  // ---- init ----
  k_partial_sum_V<<<g, blk, 0, stream>>>(V, pA, pB);
  k_finish_mean<<<1, blk, 0, stream>>>(S, pA);
  k_coeff<<<g, blk, 0, stream>>>(V, M1, M2, boo, bmo, bom, bop, bpo);
  k_init_fields<<<g, blk, 0, stream>>>(S, boo, bmo, bom, bop, bpo,
                                       x, r, r0, p, v, s, t, pA, pB);
  k_init_reduce<<<1, blk, 0, stream>>>(S, pA);

  // ---- 16 masked BiCGSTAB steps (all gating on device-side flags) ----
  for (int it = 0; it < 16; ++it) {
    k_spmv_v<<<g, blk, 0, stream>>>(S, kmaxp, boo, bmo, bom, bop, bpo, p, r0, v, pA, pB);
    k_reduce_sigma<<<1, blk, 0, stream>>>(S, kmaxp, pA, pB);
    k_spmv_restart<<<g, blk, 0, stream>>>(S, boo, bmo, bom, bop, bpo, x, p, r, r0, pA, pB);
    k_reduce_restart<<<1, blk, 0, stream>>>(S, pA);
    k_axpy_s<<<g, blk, 0, stream>>>(S, r, v, r0, s, pA, pB);
    k_reduce_s<<<1, blk, 0, stream>>>(S, pA, pB);
    k_update_c3<<<g, blk, 0, stream>>>(S, p, s, x, r);
    k_spmv_t<<<g, blk, 0, stream>>>(S, boo, bmo, bom, bop, bpo, s, t, pA, pB);
    k_reduce_omega<<<1, blk, 0, stream>>>(S, pA, pB);
    k_update_c4<<<g, blk, 0, stream>>>(S, p, s, t, r0, x, r, pA, pB);
    k_reduce_beta<<<1, blk, 0, stream>>>(S, pA, pB);
    k_update_p<<<g, blk, 0, stream>>>(S, r, v, p);
  }

  // ---- output: 8 x 514 x 514 ----
  const long long TOT = (long long)NCH * OUT2;
  const int nb = (int)((TOT + 1023) / 1024);
  k_output<<<nb, blk, 0, stream>>>(S, x, outp);
}